// GaussianOverlapLoss_16733192585480
// MI455X (gfx1250) — compile-verified
//
#include <hip/hip_runtime.h>
#include <hip/hip_bf16.h>

// GaussianOverlapLoss for MI455X (gfx1250, wave32).
// Pairwise numerator/denominator are low-rank bilinear forms in per-point
// features -> computed per 16x16 tile with V_WMMA_F32_16X16X4_F32 (f32 WMMA,
// required for range: features reach ~1e6, f16/bf16 unusable). Elementwise
// exp/sqrt/leaky_relu uses raw trans ops (v_rcp/v_sqrt/v_exp_f32) which
// co-execute with the matrix pipe; data is L2-resident so the kernel is
// trans/VALU-bound, not memory-bound.

#define NPTS   4096
#define EPSV   1e-7f
#define ALPHAV 0.01f
#define BETAV  0.6065f
#define LAMBV  1e-4f
#define NTILE  (NPTS / 16)              // 256
#define NTILES (NTILE * NTILE)          // 65536 16x16 tiles
#define WAVES_PER_BLOCK 8
#define PAIR_BLOCKS (NTILES / WAVES_PER_BLOCK)  // 8192
#define LOG2E  1.4426950408889634f

typedef float v2f __attribute__((ext_vector_type(2)));
typedef float v8f __attribute__((ext_vector_type(8)));

static __device__ __forceinline__ v8f wmma_f32_k4(v2f a, v2f b, v8f c) {
  // (neg_a, A, neg_b, B, c_mod, C, reuse_a, reuse_b) — 8-arg form per probe.
  return __builtin_amdgcn_wmma_f32_16x16x4_f32(false, a, false, b, (short)0, c,
                                               false, false);
}

// ---------------------------------------------------------------------------
// Kernel 1: per-point features.
// NUM(i,j) = sum_k P16[i][k] * Q16[j][k]   (rank 16, exact)
// DEN(i,j) = sum_k Pd8[i][k] * Qd8[j][k]   (rank 6, padded to 8, exact)
// Also det_s per point and the log1p(sqrt(eig)) terms for the lambda loss.
// ---------------------------------------------------------------------------
__global__ void gol_features(const float* __restrict__ xy,
                             const float* __restrict__ sigma,
                             float* __restrict__ P16, float* __restrict__ Q16,
                             float* __restrict__ Pd8, float* __restrict__ Qd8,
                             float* __restrict__ dets,
                             float* __restrict__ lambpart) {
  int n = blockIdx.x * blockDim.x + threadIdx.x;
  if (n >= NPTS) return;
  float x = xy[2 * n + 0], y = xy[2 * n + 1];
  float a = sigma[4 * n + 0], b = sigma[4 * n + 1];
  float c = sigma[4 * n + 2], d = sigma[4 * n + 3];
  float e   = b + c;
  float ds  = a * d - b * c;
  float dt2 = 0.25f * ds;
  float hx2 = 0.5f * x * x, hy2 = 0.5f * y * y, hxy = 0.5f * x * y;
  float w = d * hx2 + a * hy2 - e * hxy;   // 0.5*(d x^2 + a y^2 - e x y)

  float* p = P16 + n * 16;
  p[0] = w;        p[1] = 1.f;   p[2] = d;           p[3] = hx2;
  p[4] = a;        p[5] = hy2;   p[6] = e;           p[7] = -hxy;
  p[8] = d * x;    p[9] = -x;    p[10] = a * y;      p[11] = -y;
  p[12] = 0.5f * e * x; p[13] = y; p[14] = 0.5f * e * y; p[15] = x;

  float* q = Q16 + n * 16;
  q[0] = 1.f;      q[1] = w;     q[2] = hx2;         q[3] = d;
  q[4] = hy2;      q[5] = a;     q[6] = -hxy;        q[7] = e;
  q[8] = -x;       q[9] = d * x; q[10] = -y;         q[11] = a * y;
  q[12] = y;       q[13] = 0.5f * e * x; q[14] = x;  q[15] = 0.5f * e * y;

  float* pd = Pd8 + n * 8;
  pd[0] = dt2; pd[1] = 1.f; pd[2] = 0.25f * a; pd[3] = 0.25f * d;
  pd[4] = -0.25f * b; pd[5] = -0.25f * c; pd[6] = 0.f; pd[7] = 0.f;

  float* qd = Qd8 + n * 8;
  qd[0] = 1.f; qd[1] = dt2; qd[2] = d; qd[3] = a;
  qd[4] = c;   qd[5] = b;   qd[6] = 0.f; qd[7] = 0.f;

  dets[n] = ds;

  // eigvalsh of 2x2 symmetric [[a,c],[c,d]] in closed form
  float t = 0.5f * (a + d);
  float disc = 0.25f * (a - d) * (a - d) + b * c;
  float s = __builtin_amdgcn_sqrtf(fmaxf(disc, 0.f));
  float L1 = __builtin_amdgcn_sqrtf(fmaxf(t + s, EPSV));
  float L2 = __builtin_amdgcn_sqrtf(fmaxf(t - s, EPSV));
  lambpart[n] = log1pf(L1) + log1pf(L2);
}

// ---------------------------------------------------------------------------
// Kernel 2: one wave per 16x16 (i,j) tile.
// A-frag layout (16x4 f32): lane L -> M = L&15, VGPR{0,1} -> K = 2*(L>>4)+{0,1}.
// B-frag mirrored (lane -> N, same K halves). C/D: lanes 0-15 hold
// M = vgpr, lanes 16-31 hold M = vgpr+8; N = lane&15.
// ---------------------------------------------------------------------------
__global__ __launch_bounds__(256) void gol_pairwise(
    const float* __restrict__ P16, const float* __restrict__ Q16,
    const float* __restrict__ Pd8, const float* __restrict__ Qd8,
    const float* __restrict__ dets, float* __restrict__ blockpart) {
  __shared__ float red[256];
  const int wave = threadIdx.x >> 5;
  const int lane = threadIdx.x & 31;
  const int tile = blockIdx.x * WAVES_PER_BLOCK + wave;
  const int ti = tile >> 8, tj = tile & 255;
  const int I = ti << 4, J = tj << 4;
  const int r  = lane & 15;     // M for A-frag, N for B-frag / C-frag
  const int kh = lane >> 4;     // K-half for A/B frags, M-half for C/D

  // NUM: K=16 -> 4 chained WMMA_F32_16X16X4_F32
  const float* Pr = P16 + (I + r) * 16 + 2 * kh;
  const float* Qr = Q16 + (J + r) * 16 + 2 * kh;
  v8f accN = {0.f, 0.f, 0.f, 0.f, 0.f, 0.f, 0.f, 0.f};
#pragma unroll
  for (int s = 0; s < 4; ++s) {
    v2f av = *(const v2f*)(Pr + 4 * s);
    v2f bv = *(const v2f*)(Qr + 4 * s);
    accN = wmma_f32_k4(av, bv, accN);
  }

  // DEN: K=8 -> 2 chained WMMA
  const float* Pdr = Pd8 + (I + r) * 8 + 2 * kh;
  const float* Qdr = Qd8 + (J + r) * 8 + 2 * kh;
  v8f accD = {0.f, 0.f, 0.f, 0.f, 0.f, 0.f, 0.f, 0.f};
#pragma unroll
  for (int s = 0; s < 2; ++s) {
    v2f av = *(const v2f*)(Pdr + 4 * s);
    v2f bv = *(const v2f*)(Qdr + 4 * s);
    accD = wmma_f32_k4(av, bv, accD);
  }

  // Elementwise on the VALU + trans pipes (raw v_rcp/v_exp/v_sqrt, no
  // IEEE refinement sequences — this path is throughput-critical).
  const int j = J + r;
  const float dsj = dets[j];
  const float4 di0 = *(const float4*)(dets + I + 8 * kh);
  const float4 di1 = *(const float4*)(dets + I + 8 * kh + 4);
  const float dsi[8] = {di0.x, di0.y, di0.z, di0.w,
                        di1.x, di1.y, di1.z, di1.w};
  float partial = 0.f;
#pragma unroll
  for (int v = 0; v < 8; ++v) {
    const int i = I + 8 * kh + v;
    float num  = accN[v];
    float den  = accD[v];                        // detM > 0 (PD averages)
    float rden = __builtin_amdgcn_rcpf(den);     // v_rcp_f32
    // exp(-0.125*maha) = exp2(maha * (-0.125*log2e)) -> single v_exp_f32
    float t0 = __builtin_amdgcn_exp2f(num * rden * (-0.125f * LOG2E));
    float t1 = __builtin_amdgcn_sqrtf(fmaxf(dsi[v] * dsj, EPSV));
    float coef = t0 * __builtin_amdgcn_sqrtf(fmaxf(t1 * rden, EPSV));
    if (i == j) coef = 0.f;                      // diagonal -> loss term 0
    float u = coef - BETAV;
    partial += (u > 0.f ? u : ALPHAV * u) + BETAV * ALPHAV;
  }

  red[threadIdx.x] = partial;
  __syncthreads();
#pragma unroll
  for (int off = 128; off > 0; off >>= 1) {
    if (threadIdx.x < off) red[threadIdx.x] += red[threadIdx.x + off];
    __syncthreads();
  }
  if (threadIdx.x == 0) blockpart[blockIdx.x] = red[0];
}

// ---------------------------------------------------------------------------
// Kernel 3: deterministic final reduction (fixed summation order, no atomics).
// out = LOSS_WEIGHT * (LAMB * mean(log1p(L)) + sum(loss)/N)
//     = (sumPairs + 0.5*LAMB*sumLamb) / N
// ---------------------------------------------------------------------------
__global__ __launch_bounds__(256) void gol_finalize(
    const float* __restrict__ blockpart, const float* __restrict__ lambpart,
    float* __restrict__ out) {
  __shared__ float red[256];
  float s = 0.f;
  for (int i = threadIdx.x; i < PAIR_BLOCKS; i += 256) s += blockpart[i];
  float sl = 0.f;
  for (int i = threadIdx.x; i < NPTS; i += 256) sl += lambpart[i];
  red[threadIdx.x] = s + 0.5f * LAMBV * sl;
  __syncthreads();
#pragma unroll
  for (int off = 128; off > 0; off >>= 1) {
    if (threadIdx.x < off) red[threadIdx.x] += red[threadIdx.x + off];
    __syncthreads();
  }
  if (threadIdx.x == 0) out[0] = red[0] * (1.f / (float)NPTS);
}

extern "C" void kernel_launch(void* const* d_in, const int* in_sizes, int n_in,
                              void* d_out, int out_size, void* d_ws,
                              size_t ws_size, hipStream_t stream) {
  const float* xy    = (const float*)d_in[0];   // N x 2 f32
  const float* sigma = (const float*)d_in[1];   // N x 2 x 2 f32
  float* ws = (float*)d_ws;
  float* P16       = ws;                       // N*16
  float* Q16       = P16 + NPTS * 16;          // N*16
  float* Pd8       = Q16 + NPTS * 16;          // N*8
  float* Qd8       = Pd8 + NPTS * 8;           // N*8
  float* dets      = Qd8 + NPTS * 8;           // N
  float* blockpart = dets + NPTS;              // PAIR_BLOCKS
  float* lambpart  = blockpart + PAIR_BLOCKS;  // N
  float* out = (float*)d_out;

  gol_features<<<NPTS / 256, 256, 0, stream>>>(xy, sigma, P16, Q16, Pd8, Qd8,
                                               dets, lambpart);
  gol_pairwise<<<PAIR_BLOCKS, 256, 0, stream>>>(P16, Q16, Pd8, Qd8, dets,
                                                blockpart);
  gol_finalize<<<1, 256, 0, stream>>>(blockpart, lambpart, out);
}